// EdgeConv_66924180406878
// MI455X (gfx1250) — compile-verified
//
#include <hip/hip_runtime.h>
#include <hip/hip_bf16.h>

typedef __attribute__((ext_vector_type(2))) float v2f;
typedef __attribute__((ext_vector_type(8))) float v8f;
typedef __attribute__((ext_vector_type(4))) unsigned int v4u;
typedef __attribute__((ext_vector_type(4))) int v4i;
typedef __attribute__((ext_vector_type(8))) int v8i;

#define BATCH 8
#define CH    64
#define NPTS  4096
#define OCH   64
#define KNN   16
#define NEG_SLOPE 0.2f
#define BN_EPS 1e-5f

#define PANEL_COLS 128          // columns staged per TDM transfer
#define PANEL_PAD  136          // 128 data dwords + 8 pad dwords per row

static __device__ __forceinline__ v8f wmma_f32(v2f a, v2f b, v8f c) {
    // D = A(16x4) * B(4x16) + C(16x16), all f32, wave32
    return __builtin_amdgcn_wmma_f32_16x16x4_f32(false, a, false, b, (short)0, c,
                                                 false, false);
}

// Issue one TDM transfer: 64 rows x 128 cols of f32, row stride 4096 elements,
// LDS-padded +8 dwords per 128 dwords (row stride 136 dwords in LDS).
static __device__ __forceinline__ void tdm_issue_panel(unsigned long long gaddr,
                                                       unsigned lds_off) {
  v4u g0;
  g0.x = 1u;                                            // count=1, no gather
  g0.y = lds_off;                                       // LDS byte address
  g0.z = (unsigned)(gaddr & 0xFFFFFFFFu);               // global_addr[31:0]
  g0.w = (unsigned)((gaddr >> 32) & 0x01FFFFFFu)        // global_addr[56:32]
         | (2u << 30);                                  // type=2 ("image")
  v8i g1;
  g1[0] = (2 << 16)            // data_size = 4 bytes
        | (1 << 20)            // pad_enable
        | (6 << 22)            // pad_interval: every 128 dwords
        | (7 << 25);           // pad_amount: 8 dwords
  g1[1] = (NPTS & 0xFFFF) << 16;            // tensor_dim0[15:0] @ bits 63:48
  g1[2] = (NPTS >> 16) | (CH << 16);        // tensor_dim0 hi | tensor_dim1 lo
  g1[3] = (PANEL_COLS << 16);               // tensor_dim1 hi | tile_dim0
  g1[4] = CH;                               // tile_dim1 = 64, tile_dim2 = 0
  g1[5] = NPTS;                             // tensor_dim0_stride[31:0]
  g1[6] = 0;                                // stride hi | tensor_dim1_stride lo
  g1[7] = 0;
  v4i z4 = {0, 0, 0, 0};
#if defined(__clang_major__) && (__clang_major__ >= 23)
  v8i z8 = {0, 0, 0, 0, 0, 0, 0, 0};
  __builtin_amdgcn_tensor_load_to_lds(g0, g1, z4, z4, z8, 0);
#else
  __builtin_amdgcn_tensor_load_to_lds(g0, g1, z4, z4, 0);
#endif
}

// ---------------------------------------------------------------- norms ----
__global__ __launch_bounds__(256) void ec_norms_kernel(
    const float* __restrict__ x, float* __restrict__ sq) {
  int i = blockIdx.x * 256 + threadIdx.x;          // 0 .. B*N-1
  int b = i >> 12;
  int n = i & (NPTS - 1);
  const float* X = x + (size_t)b * CH * NPTS + n;
  float s = 0.f;
#pragma unroll
  for (int c = 0; c < CH; ++c) {
    float v = X[(size_t)c * NPTS];
    s = fmaf(v, v, s);
  }
  sq[i] = s;
}

// ------------------------------------------------------------------ knn ----
// One wave owns 16 query rows. Column panels (128 cols x 64 ch) are staged
// into LDS by the Tensor Data Mover (double-buffered, issued by wave 0 only;
// TDM ignores EXEC so the issue is branch-guarded at wave granularity) and
// shared by all 8 waves. Rank value = sq[col] - 2*dot (row term dropped:
// ranking-invariant).
__global__ __launch_bounds__(256) void ec_knn_kernel(
    const float* __restrict__ x, const float* __restrict__ sq,
    int* __restrict__ idxOut) {
  __shared__ float xs[2][CH][PANEL_PAD];           // 2 x 34 KB TDM panels
  __shared__ float tile[8][16 * 17];               // per-wave dist tiles
  const int t = threadIdx.x;
  const int wave = t >> 5, lane = t & 31;
  const int hi = lane >> 4, l = lane & 15;
  const int b = blockIdx.x >> 5;                   // 32 row-chunks per batch
  const int chunk = blockIdx.x & 31;
  const int q0 = chunk * 128 + wave * 16;
  const float* X = x + (size_t)b * CH * NPTS;
  const float* SQ = sq + (size_t)b * NPTS;
  const unsigned long long gbase = (unsigned long long)(const void*)X;

  // Cache all A fragments (16 k-steps of 4) in VGPRs for the whole kernel.
  v2f afrag[16];
#pragma unroll
  for (int s = 0; s < 16; ++s) {
    int c = 4 * s + 2 * hi;
    afrag[s].x = X[(size_t)c * NPTS + q0 + l];
    afrag[s].y = X[(size_t)(c + 1) * NPTS + q0 + l];
  }

  float best[16];
  int bidx[16];
#pragma unroll
  for (int j = 0; j < 16; ++j) { best[j] = 3.4e38f; bidx[j] = 0; }

  float* myTile = &tile[wave][0];

  if (wave == 0)                                   // prologue: panel 0
    tdm_issue_panel(gbase, (unsigned)(size_t)&xs[0][0][0]);

  for (int cc = 0; cc < NPTS / PANEL_COLS; ++cc) {
    if (wave == 0) {
      if (cc + 1 < NPTS / PANEL_COLS) {
        tdm_issue_panel(gbase + (unsigned long long)(cc + 1) * PANEL_COLS * 4,
                        (unsigned)(size_t)&xs[(cc + 1) & 1][0][0]);
        __builtin_amdgcn_s_wait_tensorcnt(1);      // panel cc complete
      } else {
        __builtin_amdgcn_s_wait_tensorcnt(0);
      }
    }
    __syncthreads();                               // panel cc visible to all
    asm volatile("" ::: "memory");
    const float* xc = &xs[cc & 1][0][0];

    for (int tt = 0; tt < PANEL_COLS / 16; ++tt) {
      const int cl = tt * 16;
      const int col0 = cc * PANEL_COLS + cl;
      v8f acc = {0.f, 0.f, 0.f, 0.f, 0.f, 0.f, 0.f, 0.f};
#pragma unroll
      for (int s = 0; s < 16; ++s) {
        int c = 4 * s + 2 * hi;
        v2f bfrag;
        bfrag.x = xc[c * PANEL_PAD + cl + l];
        bfrag.y = xc[(c + 1) * PANEL_PAD + cl + l];
        acc = wmma_f32(afrag[s], bfrag, acc);
      }
      const float sqc = SQ[col0 + l];
#pragma unroll
      for (int r = 0; r < 8; ++r) {
        int m = r + 8 * hi;
        myTile[m * 17 + l] = fmaf(-2.0f, acc[r], sqc);
      }
      __builtin_amdgcn_wave_barrier();             // wave-local LDS RAW order

      if (hi == 0) {                               // lane l scans row l
        const float* rowp = &myTile[l * 17];
#pragma unroll
        for (int n = 0; n < 16; ++n) {
          float dv = rowp[n];
          if (dv < best[15]) {                     // rare; stable tie-break (<)
            int iv = col0 + n;
            int p = 0;
#pragma unroll
            for (int j = 0; j < 16; ++j) p += (best[j] <= dv) ? 1 : 0;
#pragma unroll
            for (int j = 15; j >= 1; --j) {        // constant-index predicated shift
              bool sh = (j > p), ins = (j == p);
              float nb = sh ? best[j - 1] : best[j]; nb = ins ? dv : nb;
              int   ni = sh ? bidx[j - 1] : bidx[j]; ni = ins ? iv : ni;
              best[j] = nb; bidx[j] = ni;
            }
            if (p == 0) { best[0] = dv; bidx[0] = iv; }
          }
        }
      }
      __builtin_amdgcn_wave_barrier();
    }
    __syncthreads();   // all waves done with buffer cc&1 before it is refilled
  }

  if (hi == 0) {
    int* op = idxOut + ((size_t)b * NPTS + q0 + l) * KNN;
#pragma unroll
    for (int j = 0; j < 16; ++j) op[j] = bidx[j];
  }
}

// ----------------------------------------------------- projection GEMMs ----
// P1[b,n,o] = xt[b,n,:] . W[o,0:64]                (neighbor weight)
// P2[b,n,o] = xt[b,n,:] . (W[o,64:128]-W[o,0:64])  (center weight)
__global__ __launch_bounds__(256) void ec_proj_kernel(
    const float* __restrict__ x, const float* __restrict__ W,
    float* __restrict__ P1, float* __restrict__ P2) {
  __shared__ float w1s[OCH * CH];
  __shared__ float wds[OCH * CH];
  const int t = threadIdx.x;
  {
    int o = t >> 2;
    int cbase = (t & 3) * 16;
#pragma unroll
    for (int i = 0; i < 16; ++i) {
      int c = cbase + i;
      float a = W[o * (2 * CH) + c];
      float d = W[o * (2 * CH) + CH + c];
      w1s[o * CH + c] = a;
      wds[o * CH + c] = d - a;
    }
  }
  __syncthreads();

  const int wave = t >> 5, lane = t & 31;
  const int hi = lane >> 4, l = lane & 15;
  const int b = blockIdx.x >> 5;
  const int chunk = blockIdx.x & 31;
  const int n0 = chunk * 128 + wave * 16;
  const float* X = x + (size_t)b * CH * NPTS;

  v8f acc1[4], acc2[4];
#pragma unroll
  for (int ot = 0; ot < 4; ++ot) {
    acc1[ot] = (v8f){0.f, 0.f, 0.f, 0.f, 0.f, 0.f, 0.f, 0.f};
    acc2[ot] = (v8f){0.f, 0.f, 0.f, 0.f, 0.f, 0.f, 0.f, 0.f};
  }
#pragma unroll
  for (int s = 0; s < 16; ++s) {                   // K = 64 in steps of 4
    int c = 4 * s + 2 * hi;
    v2f a;
    a.x = X[(size_t)c * NPTS + n0 + l];
    a.y = X[(size_t)(c + 1) * NPTS + n0 + l];
#pragma unroll
    for (int ot = 0; ot < 4; ++ot) {
      int o = ot * 16 + l;
      v2f b1, b2;
      b1.x = w1s[o * CH + c]; b1.y = w1s[o * CH + c + 1];
      b2.x = wds[o * CH + c]; b2.y = wds[o * CH + c + 1];
      acc1[ot] = wmma_f32(a, b1, acc1[ot]);
      acc2[ot] = wmma_f32(a, b2, acc2[ot]);
    }
  }
  const size_t base = (size_t)b * NPTS + n0;
#pragma unroll
  for (int ot = 0; ot < 4; ++ot) {
#pragma unroll
    for (int r = 0; r < 8; ++r) {
      int m = r + 8 * hi;
      int o = ot * 16 + l;
      P1[(base + m) * OCH + o] = acc1[ot][r];
      P2[(base + m) * OCH + o] = acc2[ot][r];
    }
  }
}

// -------------------------------------------------------------- BN stats ----
// y[b,o,n,k] = P1[b,idx[b,n,k],o] + P2[b,n,o]; per-channel sum & sumsq.
// Deterministic: wave-local accum -> LDS -> per-block partials (no atomics).
__global__ __launch_bounds__(256) void ec_stats_kernel(
    const float* __restrict__ P1, const float* __restrict__ P2,
    const int* __restrict__ idx, float* __restrict__ partS,
    float* __restrict__ partQ) {
  __shared__ float redS[8][64];
  __shared__ float redQ[8][64];
  const int t = threadIdx.x, wave = t >> 5, lane = t & 31;
  float s1 = 0.f, q1 = 0.f, s2 = 0.f, q2 = 0.f;
  const int itemBase = blockIdx.x * 512 + wave * 64;
  for (int it = 0; it < 64; ++it) {
    const int bn = itemBase + it;
    const int b = bn >> 12;
    const float* p2row = P2 + (size_t)bn * OCH;
    const float pa = p2row[lane];
    const float pb = p2row[lane + 32];
    const int* ip = idx + (size_t)bn * KNN;
#pragma unroll
    for (int k = 0; k < KNN; ++k) {
      int id = ip[k];
      const float* r = P1 + ((size_t)b * NPTS + id) * OCH;
      float y1 = r[lane] + pa;
      float y2 = r[lane + 32] + pb;
      s1 += y1; q1 = fmaf(y1, y1, q1);
      s2 += y2; q2 = fmaf(y2, y2, q2);
    }
  }
  redS[wave][lane] = s1; redS[wave][lane + 32] = s2;
  redQ[wave][lane] = q1; redQ[wave][lane + 32] = q2;
  __syncthreads();
  if (t < 64) {
    float S = 0.f, Q = 0.f;
#pragma unroll
    for (int w = 0; w < 8; ++w) { S += redS[w][t]; Q += redQ[w][t]; }
    partS[blockIdx.x * 64 + t] = S;
    partQ[blockIdx.x * 64 + t] = Q;
  }
}

__global__ __launch_bounds__(64) void ec_finalize_stats_kernel(
    const float* __restrict__ partS, const float* __restrict__ partQ,
    const float* __restrict__ gamma, const float* __restrict__ beta,
    float* __restrict__ scale, float* __restrict__ shift) {
  const int o = threadIdx.x;
  float S = 0.f, Q = 0.f;
  for (int i = 0; i < 64; ++i) { S += partS[i * 64 + o]; Q += partQ[i * 64 + o]; }
  const float inv = 1.0f / (float)(BATCH * NPTS * KNN);
  float mean = S * inv;
  float var = fmaf(Q, inv, -mean * mean);
  float rs = rsqrtf(var + BN_EPS);
  float sc = gamma[o] * rs;
  scale[o] = sc;
  shift[o] = beta[o] - mean * sc;
}

// ---------------------------------------------------------------- output ----
// Affine + LeakyReLU are monotone (direction given by sign of scale), so
// max_k leaky(scale*y+shift) = leaky(scale * (scale>=0 ? max_k y : min_k y) + shift).
__global__ __launch_bounds__(256) void ec_output_kernel(
    const float* __restrict__ P1, const float* __restrict__ P2,
    const int* __restrict__ idx, const float* __restrict__ scale,
    const float* __restrict__ shift, float* __restrict__ out) {
  __shared__ float ych[64][33];                    // [o][n-local], padded
  const int t = threadIdx.x, wave = t >> 5, lane = t & 31;
  const int b = blockIdx.x >> 7;                   // 128 n-chunks of 32
  const int n0 = (blockIdx.x & 127) * 32;
  const float sc1 = scale[lane], sh1 = shift[lane];
  const float sc2 = scale[lane + 32], sh2 = shift[lane + 32];
#pragma unroll
  for (int j = 0; j < 4; ++j) {
    const int nl = wave * 4 + j;
    const int bn = b * NPTS + n0 + nl;
    const float* p2row = P2 + (size_t)bn * OCH;
    const float pa = p2row[lane];
    const float pb = p2row[lane + 32];
    const int* ip = idx + (size_t)bn * KNN;
    float hi1 = -3.4e38f, lo1 = 3.4e38f, hi2 = -3.4e38f, lo2 = 3.4e38f;
#pragma unroll
    for (int k = 0; k < KNN; ++k) {
      int id = ip[k];
      const float* r = P1 + ((size_t)b * NPTS + id) * OCH;
      float y1 = r[lane] + pa;
      float y2 = r[lane + 32] + pb;
      hi1 = fmaxf(hi1, y1); lo1 = fminf(lo1, y1);
      hi2 = fmaxf(hi2, y2); lo2 = fminf(lo2, y2);
    }
    float ya = (sc1 >= 0.f) ? hi1 : lo1;
    float yb = (sc2 >= 0.f) ? hi2 : lo2;
    float v1 = fmaf(sc1, ya, sh1); v1 = (v1 >= 0.f) ? v1 : NEG_SLOPE * v1;
    float v2 = fmaf(sc2, yb, sh2); v2 = (v2 >= 0.f) ? v2 : NEG_SLOPE * v2;
    ych[lane][nl] = v1;
    ych[lane + 32][nl] = v2;
  }
  __syncthreads();
#pragma unroll
  for (int j = 0; j < 8; ++j) {                    // coalesced (B,O,N) stores
    int o = (t >> 5) + j * 8;
    int nl = t & 31;
    out[((size_t)(b * OCH + o)) * NPTS + n0 + nl] = ych[o][nl];
  }
}

// ------------------------------------------------------------------ host ----
extern "C" void kernel_launch(void* const* d_in, const int* in_sizes, int n_in,
                              void* d_out, int out_size, void* d_ws, size_t ws_size,
                              hipStream_t stream) {
  const float* x     = (const float*)d_in[0];   // (8,64,4096)
  const float* W     = (const float*)d_in[1];   // (64,128)
  const float* gamma = (const float*)d_in[2];   // (64)
  const float* beta  = (const float*)d_in[3];   // (64)
  float* out = (float*)d_out;                   // (8,64,4096)

  char* ws = (char*)d_ws;
  constexpr size_t off_sq    = 0;                                 // 32768 f
  constexpr size_t off_idx   = off_sq + (size_t)BATCH * NPTS * 4; // B*N*K i32
  constexpr size_t off_P1    = off_idx + (size_t)BATCH * NPTS * KNN * 4;
  constexpr size_t off_P2    = off_P1 + (size_t)BATCH * NPTS * OCH * 4;
  constexpr size_t off_partS = off_P2 + (size_t)BATCH * NPTS * OCH * 4;
  constexpr size_t off_partQ = off_partS + 64 * 64 * 4;
  constexpr size_t off_scale = off_partQ + 64 * 64 * 4;
  constexpr size_t off_shift = off_scale + 64 * 4;

  float* sq    = (float*)(ws + off_sq);
  int*   idx   = (int*)(ws + off_idx);
  float* P1    = (float*)(ws + off_P1);
  float* P2    = (float*)(ws + off_P2);
  float* partS = (float*)(ws + off_partS);
  float* partQ = (float*)(ws + off_partQ);
  float* scale = (float*)(ws + off_scale);
  float* shift = (float*)(ws + off_shift);

  ec_norms_kernel<<<dim3(BATCH * NPTS / 256), dim3(256), 0, stream>>>(x, sq);
  ec_knn_kernel<<<dim3(BATCH * 32), dim3(256), 0, stream>>>(x, sq, idx);
  ec_proj_kernel<<<dim3(BATCH * 32), dim3(256), 0, stream>>>(x, W, P1, P2);
  ec_stats_kernel<<<dim3(64), dim3(256), 0, stream>>>(P1, P2, idx, partS, partQ);
  ec_finalize_stats_kernel<<<dim3(1), dim3(64), 0, stream>>>(partS, partQ, gamma,
                                                             beta, scale, shift);
  ec_output_kernel<<<dim3(BATCH * 128), dim3(256), 0, stream>>>(P1, P2, idx, scale,
                                                                shift, out);
}